// Cosine_SCCL_41523743817789
// MI455X (gfx1250) — compile-verified
//
#include <hip/hip_runtime.h>
#include <math.h>

typedef __attribute__((ext_vector_type(2))) float v2f;
typedef __attribute__((ext_vector_type(4))) float v4f;
typedef __attribute__((ext_vector_type(8))) float v8f;

#define N_SAMPLES 16384
#define FEAT 2048
#define NCLS 1000

// workspace layout (float offsets)
#define OFF_MEANS   0                        // NCLS*FEAT : class sums -> class means (in place)
#define OFF_COUNTS  (NCLS * FEAT)            // NCLS ints
#define OFF_OVERALL (OFF_COUNTS + NCLS)      // FEAT
#define OFF_NORMC   (OFF_OVERALL + FEAT)     // NCLS
#define OFF_SCAL    (OFF_NORMC + NCLS)       // 8 : [0]=||m||, [1]=st accumulator
#define OFF_PC      (OFF_SCAL + 8)           // NCLS : per-class sum of d^2
#define WS_FLOATS   (OFF_PC + NCLS)          // ~2.05M floats (~8.2 MB)

__global__ void k_zero(float* __restrict__ ws) {
    int i = blockIdx.x * blockDim.x + threadIdx.x;
    if (i < WS_FLOATS) ws[i] = 0.0f;
}

// One block per sample: scatter-add feature row into per-class sum, bump count.
__global__ void k_scatter(const float* __restrict__ X, const int* __restrict__ lab,
                          float* __restrict__ S, int* __restrict__ counts) {
    int s = blockIdx.x;
    int l = lab[s];
    const float* x = X + (size_t)s * FEAT;
    float* Sr = S + (size_t)l * FEAT;
    int base = threadIdx.x * 8;
#pragma unroll
    for (int j = 0; j < 8; ++j)
        atomicAdd(&Sr[base + j], x[base + j]);
    if (threadIdx.x == 0) atomicAdd(&counts[l], 1);
}

// Per-dim: overall mean = sum over class sums / N ; class sums -> class means in place.
__global__ void k_means(float* __restrict__ S, const int* __restrict__ counts,
                        float* __restrict__ overall) {
    int d = blockIdx.x * blockDim.x + threadIdx.x;   // 0..FEAT-1
    float sum = 0.0f;
    for (int c = 0; c < NCLS; ++c) {
        float v = S[(size_t)c * FEAT + d];
        sum += v;
        float cnt = (float)counts[c];
        S[(size_t)c * FEAT + d] = (cnt > 0.0f) ? (v / cnt) : 0.0f;  // nan_to_num
    }
    overall[d] = sum / (float)N_SAMPLES;
}

// Block c < NCLS: ||mean_c||; block NCLS: ||overall||.
__global__ void k_norms(const float* __restrict__ S, const float* __restrict__ overall,
                        float* __restrict__ normc, float* __restrict__ scal) {
    __shared__ float red[256];
    int c = blockIdx.x;
    const float* r = (c < NCLS) ? (S + (size_t)c * FEAT) : overall;
    float acc = 0.0f;
    for (int d = threadIdx.x; d < FEAT; d += 256) { float v = r[d]; acc += v * v; }
    red[threadIdx.x] = acc;
    __syncthreads();
    for (int off = 128; off > 0; off >>= 1) {
        if (threadIdx.x < off) red[threadIdx.x] += red[threadIdx.x + off];
        __syncthreads();
    }
    if (threadIdx.x == 0) {
        float n = sqrtf(red[0]);
        if (c < NCLS) normc[c] = n; else scal[0] = n;
    }
}

// Main pass: 1 wave = 16-sample tile. WMMA f32 16x16x4:
//   C1 = X_tile * B   (B col j = mean of label of sample j)  -> diag = x_i . mu_{l_i}
//   C2 = X_tile * Bm  (Bm col j = overall mean)              -> any col = x_i . m
__global__ __launch_bounds__(128) void k_main(
    const float* __restrict__ X, const int* __restrict__ lab,
    const float* __restrict__ means, const float* __restrict__ overall,
    const float* __restrict__ normc, const float* __restrict__ scal,
    float* __restrict__ per_class, float* __restrict__ st_acc) {
    int lane = threadIdx.x & 31;
    int wid  = blockIdx.x * 4 + (threadIdx.x >> 5);
    int row  = lane & 15;      // sample row within tile / B column index
    int h    = lane >> 4;      // which K half this lane feeds
    int sample = wid * 16 + row;
    int l = lab[sample];
    const float* xr = X + (size_t)sample * FEAT;
    const float* mr = means + (size_t)l * FEAT;

    v8f c1 = {};
    v8f c2 = {};
    float ss = 0.0f;

    for (int k = 0; k < FEAT; k += 16) {
        int o = k + 8 * h;
        v4f xa = *(const v4f*)(xr + o);
        v4f xb = *(const v4f*)(xr + o + 4);
        v4f ma = *(const v4f*)(mr + o);
        v4f mb = *(const v4f*)(mr + o + 4);
        v4f oa = *(const v4f*)(overall + o);
        v4f ob = *(const v4f*)(overall + o + 4);

        v2f a0 = {xa.x, xa.y}, a1 = {xa.z, xa.w}, a2 = {xb.x, xb.y}, a3 = {xb.z, xb.w};
        v2f b0 = {ma.x, ma.y}, b1 = {ma.z, ma.w}, b2 = {mb.x, mb.y}, b3 = {mb.z, mb.w};
        v2f g0 = {oa.x, oa.y}, g1 = {oa.z, oa.w}, g2 = {ob.x, ob.y}, g3 = {ob.z, ob.w};

        c1 = __builtin_amdgcn_wmma_f32_16x16x4_f32(false, a0, false, b0, (short)0, c1, false, false);
        c2 = __builtin_amdgcn_wmma_f32_16x16x4_f32(false, a0, false, g0, (short)0, c2, false, false);
        c1 = __builtin_amdgcn_wmma_f32_16x16x4_f32(false, a1, false, b1, (short)0, c1, false, false);
        c2 = __builtin_amdgcn_wmma_f32_16x16x4_f32(false, a1, false, g1, (short)0, c2, false, false);
        c1 = __builtin_amdgcn_wmma_f32_16x16x4_f32(false, a2, false, b2, (short)0, c1, false, false);
        c2 = __builtin_amdgcn_wmma_f32_16x16x4_f32(false, a2, false, g2, (short)0, c2, false, false);
        c1 = __builtin_amdgcn_wmma_f32_16x16x4_f32(false, a3, false, b3, (short)0, c1, false, false);
        c2 = __builtin_amdgcn_wmma_f32_16x16x4_f32(false, a3, false, g3, (short)0, c2, false, false);

        ss += xa.x * xa.x + xa.y * xa.y + xa.z * xa.z + xa.w * xa.w
            + xb.x * xb.x + xb.y * xb.y + xb.z * xb.z + xb.w * xb.w;
    }

    // combine ||x_row||^2 halves: lanes L and L+16 covered disjoint K
    ss += __shfl_xor(ss, 16, 32);

    // Extract diag(C1) and a column of C2.
    // C layout: VGPR v, lanes 0-15: M=v, N=lane ; lanes 16-31: M=v+8, N=lane-16.
    float dotc = 0.0f, dotm = 0.0f;
#pragma unroll
    for (int i = 0; i < 16; ++i) {
        int src = (i < 8) ? i : (16 + i);
        float v1 = __shfl(c1[i & 7], src, 32);
        float v2 = __shfl(c2[i & 7], src, 32);
        if (lane == i) { dotc = v1; dotm = v2; }
    }

    float nx = sqrtf(ss);
    float nm = scal[0];
    float nc = normc[l];
    // faithful precedence: 1 - (dot/||x||) * ||v||
    float d1 = 1.0f - (dotm / nx) * nm;
    float d2 = 1.0f - (dotc / nx) * nc;

    if (lane < 16) atomicAdd(&per_class[l], d2 * d2);

    float s1 = (lane < 16) ? d1 * d1 : 0.0f;
    s1 += __shfl_xor(s1, 16, 32);
    s1 += __shfl_xor(s1, 8, 32);
    s1 += __shfl_xor(s1, 4, 32);
    s1 += __shfl_xor(s1, 2, 32);
    s1 += __shfl_xor(s1, 1, 32);
    if (lane == 0) atomicAdd(st_acc, s1);
}

__global__ void k_final(const int* __restrict__ counts, const float* __restrict__ per_class,
                        const float* __restrict__ scal, float* __restrict__ out) {
    __shared__ float red[256];
    float acc = 0.0f;
    for (int c = threadIdx.x; c < NCLS; c += 256) {
        float cnt = (float)counts[c];
        // faithful to (1/Ni) * per_class * (Ni/N)
        acc += (1.0f / cnt) * per_class[c] * (cnt / (float)N_SAMPLES);
    }
    red[threadIdx.x] = acc;
    __syncthreads();
    for (int off = 128; off > 0; off >>= 1) {
        if (threadIdx.x < off) red[threadIdx.x] += red[threadIdx.x + off];
        __syncthreads();
    }
    if (threadIdx.x == 0) {
        float st = scal[1] / (float)N_SAMPLES;
        float sw = red[0] / (float)NCLS;
        out[0] = st;
        out[1] = sw;
        out[2] = st - sw;
    }
}

extern "C" void kernel_launch(void* const* d_in, const int* in_sizes, int n_in,
                              void* d_out, int out_size, void* d_ws, size_t ws_size,
                              hipStream_t stream) {
    (void)in_sizes; (void)n_in; (void)out_size; (void)ws_size;
    const float* X  = (const float*)d_in[0];
    const int* lab  = (const int*)d_in[1];
    float* out      = (float*)d_out;

    float* ws      = (float*)d_ws;
    float* S       = ws + OFF_MEANS;
    int*   counts  = (int*)(ws + OFF_COUNTS);
    float* overall = ws + OFF_OVERALL;
    float* normc   = ws + OFF_NORMC;
    float* scal    = ws + OFF_SCAL;
    float* pc      = ws + OFF_PC;

    k_zero<<<(WS_FLOATS + 255) / 256, 256, 0, stream>>>(ws);
    k_scatter<<<N_SAMPLES, 256, 0, stream>>>(X, lab, S, counts);
    k_means<<<FEAT / 256, 256, 0, stream>>>(S, counts, overall);
    k_norms<<<NCLS + 1, 256, 0, stream>>>(S, overall, normc, scal);
    k_main<<<N_SAMPLES / 64, 128, 0, stream>>>(X, lab, S, overall, normc, scal,
                                               pc, scal + 1);
    k_final<<<1, 256, 0, stream>>>(counts, pc, scal, out);
}